// MyChamferLoss_87625922773776
// MI455X (gfx1250) — compile-verified
//
#include <hip/hip_runtime.h>

typedef __attribute__((ext_vector_type(2))) float v2f;
typedef __attribute__((ext_vector_type(8))) float v8f;

#define B_ 8
#define N_ 4096
#define M_ 4096
#define ROWS_PER_BLOCK 128   // 8 wave32s x 16 pred rows
#define THREADS 256
#define INF_BITS 0x7F800000u

// ---------------------------------------------------------------------------
// Kernel 0: initialize min buffers to +inf (must run every call: the harness
// does not re-poison workspace between graph replays).
// ---------------------------------------------------------------------------
__global__ void chamfer_init(unsigned* __restrict__ colmin,
                             float* __restrict__ rowmin) {
    int i = blockIdx.x * blockDim.x + threadIdx.x;
    if (i < B_ * M_) colmin[i] = INF_BITS;
    if (i < B_ * N_) rowmin[i] = __builtin_inff();
}

// ---------------------------------------------------------------------------
// Kernel 1: tiled pairwise SQUARED distances via V_WMMA_F32_16X16X4_F32.
// Full K=4 utilization -- the WMMA emits d^2 directly:
//   A (16x4)  = (-2px, -2py, -2pz, 1)       per pred row
//   B (4x16)  = ( gx ,  gy ,  gz , ||g||^2) per gt col
//   C (16x16) = ||p||^2 broadcast per row
//   D = ||p||^2 - 2 p.g + ||g||^2 = d^2
// Mins tracked in d^2 space (sqrt is monotone); sqrt deferred to the end.
// Hot loop is straight-line, full-EXEC: B operand software-pipelined; both
// lane halves push their partial column min via ds_min_u32 (no bpermute, no
// exec-mask branch).
// ---------------------------------------------------------------------------
__global__ __launch_bounds__(THREADS)
void chamfer_tiles(const float* __restrict__ pred,
                   const float* __restrict__ gt,
                   unsigned* __restrict__ colmin,
                   float* __restrict__ rowmin) {
    extern __shared__ float smem[];
    float*    sgt4 = smem;                          // M_*4 floats (64 KB)
    unsigned* scol = (unsigned*)(sgt4 + M_ * 4);    // M_   uints  (16 KB)

    const int b   = blockIdx.y;
    const int tid = threadIdx.x;

    // Stage gt batch into LDS as {x, y, z, ||g||^2} quads; init LDS col-min.
    const float* gtb = gt + (size_t)b * M_ * 3;
    for (int i = tid; i < M_; i += THREADS) {
        float x = gtb[i * 3 + 0];
        float y = gtb[i * 3 + 1];
        float z = gtb[i * 3 + 2];
        sgt4[i * 4 + 0] = x;
        sgt4[i * 4 + 1] = y;
        sgt4[i * 4 + 2] = z;
        sgt4[i * 4 + 3] = x * x + y * y + z * z;
        scol[i] = INF_BITS;
    }
    __syncthreads();

    const int wv    = tid >> 5;
    const int lane  = tid & 31;
    const int half  = lane >> 4;        // 0: K=0,1 ; 1: K=2,3
    const int l15   = lane & 15;
    const int rbase = blockIdx.x * ROWS_PER_BLOCK + wv * 16;

    // A operand (16x4 f32): lanes 0-15 -> K=(-2px,-2py); 16-31 -> K=(-2pz,1).
    const float* pr = pred + ((size_t)b * N_ + rbase + l15) * 3;
    float px = pr[0], py = pr[1], pz = pr[2];
    float p2 = px * px + py * py + pz * pz;
    v2f a;
    a.x = -2.0f * (half ? pz : px);
    a.y = half ? 1.0f : (-2.0f * py);

    // C seed: VGPR v needs ||p||^2 of row rbase+v (lanes 0-15) / rbase+v+8
    // (lanes 16-31). Lane v holds p2 of row rbase+v.
    v8f cinit;
#pragma unroll
    for (int v = 0; v < 8; ++v)
        cinit[v] = __shfl(p2, v + (half ? 8 : 0), 32);

    v8f rmin;
#pragma unroll
    for (int v = 0; v < 8; ++v) rmin[v] = __builtin_inff();

    // B operand: one ds_load_b64 per lane per tile, conflict-free, software
    // pipelined one tile ahead so the min-tree + atomic hide its latency.
    const v2f* bptr = (const v2f*)sgt4 + (size_t)l15 * 2 + half;
    unsigned* scolme = scol + l15;      // this lane's column slot (per tile)

    v2f bv = bptr[0];
#pragma unroll 2
    for (int ct = 0; ct < M_ / 16; ++ct) {
        v2f bnext = (ct + 1 < M_ / 16) ? bptr[(ct + 1) * 32] : bv;

        v8f acc = __builtin_amdgcn_wmma_f32_16x16x4_f32(
            /*neg_a=*/false, a, /*neg_b=*/false, bv,
            /*c_mod=*/(short)0, cinit, /*reuse_a=*/false, /*reuse_b=*/false);

        // Column partial min: tree over this lane-half's 8 d^2 values.
        float m01 = fminf(acc[0], acc[1]);
        float m23 = fminf(acc[2], acc[3]);
        float m45 = fminf(acc[4], acc[5]);
        float m67 = fminf(acc[6], acc[7]);
        float cmin = fminf(fminf(m01, m23), fminf(m45, m67));

        // Row mins accumulate in registers.
#pragma unroll
        for (int v = 0; v < 8; ++v) rmin[v] = fminf(rmin[v], acc[v]);

        // Both halves hit scol[col] (lanes L and L+16 share the address);
        // the LDS atomic unit merges them. Clamp so uint ordering is valid.
        cmin = fmaxf(cmin, 0.0f);
        atomicMin(&scolme[ct * 16], __float_as_uint(cmin));  // ds_min_u32

        bv = bnext;
    }

    // Row mins: reduce across the 16 lanes of each half (masks stay in-half).
#pragma unroll
    for (int m = 8; m >= 1; m >>= 1) {
#pragma unroll
        for (int v = 0; v < 8; ++v)
            rmin[v] = fminf(rmin[v], __shfl_xor(rmin[v], m, 32));
    }
    if (l15 == 0) {
        float* rout = rowmin + (size_t)b * N_ + rbase + half * 8;
#pragma unroll
        for (int v = 0; v < 8; ++v)
            rout[v] = __builtin_sqrtf(fmaxf(rmin[v], 0.0f));
    }

    __syncthreads();
    // Flush per-workgroup column mins (still d^2 bits) to global.
    unsigned* cg = colmin + (size_t)b * M_;
    for (int i = tid; i < M_; i += THREADS)
        atomicMin(&cg[i], scol[i]);
}

// ---------------------------------------------------------------------------
// Kernel 2: final scalar = mean(sqrt(colmin_d2)) + mean(rowmin_dist).
// ---------------------------------------------------------------------------
__global__ __launch_bounds__(256)
void chamfer_finalize(const unsigned* __restrict__ colmin,
                      const float* __restrict__ rowmin,
                      float* __restrict__ out) {
    __shared__ float s1[256];
    __shared__ float s2[256];
    int tid = threadIdx.x;
    float a = 0.0f, r = 0.0f;
    for (int i = tid; i < B_ * M_; i += 256)
        a += __builtin_sqrtf(__uint_as_float(colmin[i]));
    for (int i = tid; i < B_ * N_; i += 256)
        r += rowmin[i];
    s1[tid] = a;
    s2[tid] = r;
    __syncthreads();
    for (int s = 128; s > 0; s >>= 1) {
        if (tid < s) {
            s1[tid] += s1[tid + s];
            s2[tid] += s2[tid + s];
        }
        __syncthreads();
    }
    if (tid == 0)
        out[0] = s1[0] / (float)(B_ * M_) + s2[0] / (float)(B_ * N_);
}

// ---------------------------------------------------------------------------
extern "C" void kernel_launch(void* const* d_in, const int* in_sizes, int n_in,
                              void* d_out, int out_size, void* d_ws,
                              size_t ws_size, hipStream_t stream) {
    const float* pred = (const float*)d_in[0];  // [B, N, 3] f32
    const float* gt   = (const float*)d_in[1];  // [B, M, 3] f32
    float* out = (float*)d_out;

    unsigned* colmin = (unsigned*)d_ws;                       // B*M uints
    float*    rowmin = (float*)((char*)d_ws +
                                (size_t)B_ * M_ * sizeof(unsigned));  // B*N f32

    chamfer_init<<<(B_ * M_ + 255) / 256, 256, 0, stream>>>(colmin, rowmin);

    size_t lds_bytes = (size_t)M_ * 4 * sizeof(float)
                     + (size_t)M_ * sizeof(unsigned);          // 80 KB
    dim3 grid(N_ / ROWS_PER_BLOCK, B_);
    chamfer_tiles<<<grid, THREADS, lds_bytes, stream>>>(pred, gt, colmin,
                                                        rowmin);

    chamfer_finalize<<<1, 256, 0, stream>>>(colmin, rowmin, out);
}